// SetOfSetLayer_111669149722
// MI455X (gfx1250) — compile-verified
//
#include <hip/hip_runtime.h>

// SetOfSet layer for MI455X (gfx1250, wave32).
// Big GEMM + combine done with V_WMMA_F32_16X16X4_F32 (exact f32 math).
// d_in order: values, row_idx, col_idx, W_all, b_all, W_n, b_n, W_m, b_m, W_both, b_both

typedef __attribute__((ext_vector_type(2))) float v2f;
typedef __attribute__((ext_vector_type(8))) float v8f;

#define D 128
#define WT_LD 132      // LDS leading dim for transposed W (pad -> conflict-free b64)
#define NROWS 500      // cameras (row_idx range)
#define MCOLS 50000    // points  (col_idx range)

// ---------------------------------------------------------------------------
// Pass 1: segment sums + counts via f32 atomics. One entry (512B row) per
// block iteration, 128 threads => one fully coalesced 512B load per entry.
// sum tables total ~26MB -> atomics resolve in L2 (192MB).
// ---------------------------------------------------------------------------
__global__ void k_accum(const float* __restrict__ values,
                        const int* __restrict__ row_idx,
                        const int* __restrict__ col_idx,
                        float* __restrict__ sum_rows,  // [MCOLS][128], keyed by col_idx
                        float* __restrict__ cnt_rows,  // [MCOLS]
                        float* __restrict__ sum_cols,  // [NROWS][128], keyed by row_idx
                        float* __restrict__ cnt_cols,  // [NROWS]
                        int nnz) {
  const int d = threadIdx.x;  // 0..127
  for (int e = blockIdx.x; e < nnz; e += gridDim.x) {
    const int c = col_idx[e];
    const int r = row_idx[e];
    const float v = values[(size_t)e * D + d];
    atomicAdd(&sum_rows[(size_t)c * D + d], v);
    atomicAdd(&sum_cols[(size_t)r * D + d], v);
    if (d == 0) {
      atomicAdd(&cnt_rows[c], 1.0f);
      atomicAdd(&cnt_cols[r], 1.0f);
    }
  }
}

// ---------------------------------------------------------------------------
// Global-mean branch: gm = (sum over all entries)/nnz = (sum of sum_cols rows)/nnz.
// combo[n] = b_all[n] + b_both[n] + sum_k gm[k]*W_both[k][n]   (tiny: 16K MACs)
// ---------------------------------------------------------------------------
__global__ void k_combo(const float* __restrict__ sum_cols,
                        const float* __restrict__ W_both,
                        const float* __restrict__ b_both,
                        const float* __restrict__ b_all,
                        float* __restrict__ combo,
                        int nrows, float inv_nnz) {
  __shared__ float gm[D];
  const int d = threadIdx.x;  // 0..127
  float s = 0.0f;
  for (int r = 0; r < nrows; ++r) s += sum_cols[(size_t)r * D + d];
  gm[d] = s * inv_nnz;
  __syncthreads();
  float acc = b_all[d] + b_both[d];
  for (int k = 0; k < D; ++k) acc += gm[k] * W_both[(size_t)k * D + d];
  combo[d] = acc;
}

// ---------------------------------------------------------------------------
// Stage 128x128 W into LDS transposed: Wt[n*WT_LD + k] = W[k*128 + n].
// ---------------------------------------------------------------------------
__device__ __forceinline__ void stage_w_transposed(float* Wt,
                                                   const float* __restrict__ W) {
  for (int i = threadIdx.x; i < D * D; i += 256) {
    const int k = i >> 7, n = i & 127;
    Wt[n * WT_LD + k] = W[i];
  }
}

// ---------------------------------------------------------------------------
// Core: one wave computes a 16-row x 128-col tile of  A(16x128) @ W(128x128).
// A fragment (f32 16x4 layout): lane holds M = lane&15; half-wave selects
// K offset {0,2}; arow must already include the +2*half element offset.
// B fragment: single ds_load_b64 from transposed W (conflict-free, padded).
// Unroll-by-4 (not full): the scheduler pipelines the 4 global + 32 DS loads
// ahead of the 32 WMMAs inside each body, but cannot hoist the whole W tile
// into registers (round-3 full unroll caused >512 VGPRs + scratch spills).
// ---------------------------------------------------------------------------
template <bool SCALE>
__device__ __forceinline__ void wmma_rowtile_16x128(
    const float* __restrict__ arow,  // = base_row + 2*half
    float scale,                     // per-row 1/count (ignored if !SCALE)
    const float* __restrict__ Wt,    // LDS, transposed, WT_LD leading dim
    int l16, int half, v8f acc[8]) {
  const float* bb = Wt + l16 * WT_LD + 2 * half;  // + t*16*WT_LD + k per frag
#pragma unroll 4
  for (int k = 0; k < D; k += 4) {
    v2f a = *(const v2f*)(arow + k);
    if (SCALE) { a.x *= scale; a.y *= scale; }
#pragma unroll
    for (int t = 0; t < 8; ++t) {
      const v2f b = *(const v2f*)(bb + k + t * (16 * WT_LD));
      acc[t] = __builtin_amdgcn_wmma_f32_16x16x4_f32(
          false, a, false, b, (short)0, acc[t], false, false);
    }
  }
}

// ---------------------------------------------------------------------------
// WMMA f32 GEMM over a mean table: out = (sums/max(cnt,1)) @ W + bias.
// One 16-row tile per wave, 8 waves/block.
// ---------------------------------------------------------------------------
__global__ void __launch_bounds__(256)
k_mean_gemm(const float* __restrict__ sums,
            const float* __restrict__ cnts,
            const float* __restrict__ W,
            const float* __restrict__ bias,
            float* __restrict__ out,
            int rows) {
  __shared__ float Wt[D * WT_LD];  // ~67.6KB (<=320KB/WG on gfx1250)
  stage_w_transposed(Wt, W);
  __syncthreads();

  const int lane = threadIdx.x & 31;
  const int wave = threadIdx.x >> 5;
  const int half = lane >> 4;   // 0: lanes 0-15, 1: lanes 16-31
  const int l16  = lane & 15;

  float bv[8];
#pragma unroll
  for (int t = 0; t < 8; ++t) bv[t] = bias[t * 16 + l16];

  const int ntiles = (rows + 15) >> 4;
  for (int tile = blockIdx.x * 8 + wave; tile < ntiles; tile += gridDim.x * 8) {
    const int m  = tile * 16 + l16;
    const int mc = m < rows ? m : rows - 1;           // clamp for partial tile
    const float inv = 1.0f / fmaxf(cnts[mc], 1.0f);
    const float* arow = sums + (size_t)mc * D + half * 2;

    v8f acc[8] = {};
    wmma_rowtile_16x128<true>(arow, inv, Wt, l16, half, acc);

#pragma unroll
    for (int v = 0; v < 8; ++v) {
      const int mm = tile * 16 + half * 8 + v;        // C layout: M = v + 8*half
      if (mm < rows) {
        float* orow = out + (size_t)mm * D;
#pragma unroll
        for (int t = 0; t < 8; ++t) orow[t * 16 + l16] = acc[t][v] + bv[t];
      }
    }
  }
}

// ---------------------------------------------------------------------------
// Fused main pass: out = (values@W_all + combo + out_rows[col] + out_cols[row])*0.25
// Each wave: 16 entries x 128 out-dims (8 accumulator tiles, 64 VGPRs).
// Gathers/stores: per C-VGPR each half-wave touches 64 contiguous bytes of a
// single row => coalesced; gather tables are L2-resident (~26MB).
// ---------------------------------------------------------------------------
__global__ void __launch_bounds__(256)
k_fused(const float* __restrict__ values,
        const int* __restrict__ row_idx,
        const int* __restrict__ col_idx,
        const float* __restrict__ W,        // W_all
        const float* __restrict__ combo,    // b_all + out_both
        const float* __restrict__ out_rows, // [MCOLS][128]
        const float* __restrict__ out_cols, // [NROWS][128]
        float* __restrict__ out,
        int nnz) {
  __shared__ float Wt[D * WT_LD];
  stage_w_transposed(Wt, W);
  __syncthreads();

  const int lane = threadIdx.x & 31;
  const int wave = threadIdx.x >> 5;
  const int half = lane >> 4;
  const int l16  = lane & 15;

  float cbv[8];
#pragma unroll
  for (int t = 0; t < 8; ++t) cbv[t] = combo[t * 16 + l16];

  const int ngroups = nnz >> 4;  // 16 entries per wave-group
  for (int g = blockIdx.x * 8 + wave; g < ngroups; g += gridDim.x * 8) {
    const int base = g * 16;
    const float* arow = values + (size_t)(base + l16) * D + half * 2;

    int cg[8], rg[8];
#pragma unroll
    for (int v = 0; v < 8; ++v) {
      cg[v] = col_idx[base + half * 8 + v];
      rg[v] = row_idx[base + half * 8 + v];
    }

    v8f acc[8] = {};
    wmma_rowtile_16x128<false>(arow, 1.0f, Wt, l16, half, acc);

#pragma unroll
    for (int v = 0; v < 8; ++v) {
      const float* rr = out_rows + (size_t)cg[v] * D;
      const float* cc = out_cols + (size_t)rg[v] * D;
      float* orow = out + (size_t)(base + half * 8 + v) * D;
#pragma unroll
      for (int t = 0; t < 8; ++t) {
        const int n = t * 16 + l16;
        orow[n] = (acc[t][v] + cbv[t] + rr[n] + cc[n]) * 0.25f;
      }
    }
  }
}

// ---------------------------------------------------------------------------
extern "C" void kernel_launch(void* const* d_in, const int* in_sizes, int n_in,
                              void* d_out, int out_size, void* d_ws, size_t ws_size,
                              hipStream_t stream) {
  const float* values = (const float*)d_in[0];
  const int*   row_idx = (const int*)d_in[1];
  const int*   col_idx = (const int*)d_in[2];
  const float* W_all  = (const float*)d_in[3];
  const float* b_all  = (const float*)d_in[4];
  const float* W_n    = (const float*)d_in[5];
  const float* b_n    = (const float*)d_in[6];
  const float* W_m    = (const float*)d_in[7];
  const float* b_m    = (const float*)d_in[8];
  const float* W_both = (const float*)d_in[9];
  const float* b_both = (const float*)d_in[10];
  float* out = (float*)d_out;
  const int nnz = in_sizes[1];  // row_idx element count

  // Workspace layout (floats). Zero-initialized region first (single memset).
  float* ws       = (float*)d_ws;
  float* sum_rows = ws;                                  // MCOLS*128
  float* sum_cols = sum_rows + (size_t)MCOLS * D;        // NROWS*128
  float* cnt_rows = sum_cols + (size_t)NROWS * D;        // MCOLS
  float* cnt_cols = cnt_rows + MCOLS;                    // NROWS
  float* out_rows = cnt_cols + NROWS;                    // MCOLS*128
  float* out_cols = out_rows + (size_t)MCOLS * D;        // NROWS*128
  float* combo    = out_cols + (size_t)NROWS * D;        // 128

  const size_t zbytes =
      ((size_t)MCOLS * D + (size_t)NROWS * D + MCOLS + NROWS) * sizeof(float);
  hipMemsetAsync(d_ws, 0, zbytes, stream);

  k_accum<<<16384, 128, 0, stream>>>(values, row_idx, col_idx,
                                     sum_rows, cnt_rows, sum_cols, cnt_cols, nnz);

  k_combo<<<1, 128, 0, stream>>>(sum_cols, W_both, b_both, b_all, combo,
                                 NROWS, 1.0f / (float)nnz);

  const int tiles_r = (MCOLS + 15) / 16;   // 3125
  k_mean_gemm<<<(tiles_r + 7) / 8, 256, 0, stream>>>(sum_rows, cnt_rows, W_n, b_n,
                                                     out_rows, MCOLS);
  const int tiles_c = (NROWS + 15) / 16;   // 32
  k_mean_gemm<<<(tiles_c + 7) / 8, 256, 0, stream>>>(sum_cols, cnt_cols, W_m, b_m,
                                                     out_cols, NROWS);

  const int ngroups = nnz / 16;            // 62500
  k_fused<<<(ngroups + 7) / 8, 256, 0, stream>>>(values, row_idx, col_idx, W_all,
                                                 combo, out_rows, out_cols, out, nnz);
}